// Net_35768487641765
// MI455X (gfx1250) — compile-verified
//
#include <hip/hip_runtime.h>
#include <cmath>

typedef __attribute__((ext_vector_type(2))) float v2f_t;
typedef __attribute__((ext_vector_type(8))) float v8f_t;

#define HN 64
#define GC 100
#define LAYERS 8
#define APAD 68   // padded LDS row stride (floats): bank = (row*4 + k) % 64, conflict-free

// ---------------------------------------------------------------- lin0
__global__ __launch_bounds__(256) void lin0_kernel(
    const float* __restrict__ pos, const float* __restrict__ w,
    const float* __restrict__ b, float* __restrict__ x,
    float* __restrict__ x0, int n)
{
    int gid = blockIdx.x * blockDim.x + threadIdx.x;
    if (gid >= n * HN) return;
    int node = gid >> 6, c = gid & 63;
    float p0 = pos[node * 3 + 0];
    float p1 = pos[node * 3 + 1];
    float p2 = pos[node * 3 + 2];
    float v = fmaf(p0, w[0 * HN + c], fmaf(p1, w[1 * HN + c], fmaf(p2, w[2 * HN + c], b[c])));
    v = v > 0.0f ? v : 0.0f;
    x[gid] = v;
    x0[gid] = v;
}

// ---------------------------------------------------------------- h = ALPHA * x0
__global__ __launch_bounds__(256) void inith_kernel(
    const float4* __restrict__ x0, float4* __restrict__ h, int n4)
{
    int i = blockIdx.x * blockDim.x + threadIdx.x;
    if (i >= n4) return;
    float4 v = x0[i];
    float4 o;
    o.x = 0.1f * v.x; o.y = 0.1f * v.y; o.z = 0.1f * v.z; o.w = 0.1f * v.w;
    h[i] = o;
}

// ---------------------------------------------------------------- edge scatter
// h[dst] += (1-ALPHA) * x[src]; 16 threads per edge, 4 channels each.
// L2-atomic bound: x (25.6MB) and h are L2-resident on MI455X (192MB L2).
__global__ __launch_bounds__(256) void scatter_kernel(
    const float* __restrict__ x, const int* __restrict__ src,
    const int* __restrict__ dst, float* __restrict__ h, int nE)
{
    int gid = blockIdx.x * blockDim.x + threadIdx.x;
    int e = gid >> 4;
    if (e >= nE) return;
    // stream the index arrays ahead (lowers to global_prefetch_b8)
    if ((gid & 15) == 0 && e + 2048 < nE) {
        __builtin_prefetch(src + e + 2048, 0, 1);
        __builtin_prefetch(dst + e + 2048, 0, 1);
    }
    int c0 = (gid & 15) * 4;
    int s = src[e], d = dst[e];
    const float4 v = *(const float4*)(x + (size_t)s * HN + c0);
    float* hp = h + (size_t)d * HN + c0;
    atomicAdd(hp + 0, 0.9f * v.x);
    atomicAdd(hp + 1, 0.9f * v.y);
    atomicAdd(hp + 2, 0.9f * v.z);
    atomicAdd(hp + 3, 0.9f * v.w);
}

// ---------------------------------------------------------------- WMMA GEMM [rows,64]@[64,64]
// Block = 8 waves, each wave owns one 16-row tile. W is staged to LDS once per
// block (transposed, padded) and the per-wave A tile is staged with coalesced
// float4 loads; all WMMA fragments then come from conflict-free ds_load_b64.
// mode 0: out = relu((1-beta)*A + beta*(A@W));  mode 1: out = A@W + bias
__global__ __launch_bounds__(256) void gemm64_kernel(
    const float* __restrict__ A, const float* __restrict__ W,
    const float* __restrict__ bias, float* __restrict__ out,
    int n_rows, float beta, int mode)
{
    __shared__ float Wt[HN * APAD];        // Wt[n*APAD + k] = W[k*HN + n]
    __shared__ float At[8][16 * APAD];     // per-wave 16x64 tile, padded rows

    const int tid  = threadIdx.x;
    const int lane = tid & 31;
    const int wave = tid >> 5;
    const int tile = blockIdx.x * 8 + wave;
    const int row_base = tile * 16;
    const bool valid = row_base < n_rows;
    const int rb = valid ? row_base : 0;

    // ---- stage W transposed (whole block): 4096 floats, coalesced reads
    for (int i = tid; i < HN * HN; i += 256) {
        int k = i >> 6, n = i & 63;
        Wt[n * APAD + k] = W[i];
    }
    // ---- stage this wave's 16x64 A tile: 256 float4, coalesced (16B/lane)
    float* __restrict__ Atw = &At[wave][0];
    {
        const float4* __restrict__ Ag = (const float4*)(A + (size_t)rb * HN);
#pragma unroll
        for (int j = 0; j < 8; ++j) {
            int idx = j * 32 + lane;         // [0,256)
            int r = idx >> 4, c4 = idx & 15; // row, float4-col
            float4 v = Ag[r * 16 + c4];
            *(float4*)&Atw[r * APAD + c4 * 4] = v;
        }
    }
    __syncthreads();

    const int m   = lane & 15;          // A row / B col within 16
    const int klo = (lane >> 4) * 2;    // lanes 0-15: k,k+1 ; lanes 16-31: k+2,k+3

    v8f_t acc0 = {}, acc1 = {}, acc2 = {}, acc3 = {};
    const float* __restrict__ Ar  = &Atw[m * APAD];
    const float* __restrict__ Wr0 = &Wt[(0 * 16 + m) * APAD];
    const float* __restrict__ Wr1 = &Wt[(1 * 16 + m) * APAD];
    const float* __restrict__ Wr2 = &Wt[(2 * 16 + m) * APAD];
    const float* __restrict__ Wr3 = &Wt[(3 * 16 + m) * APAD];

#pragma unroll
    for (int kb = 0; kb < HN; kb += 4) {
        const int k0 = kb + klo;
        v2f_t a  = *(const v2f_t*)&Ar[k0];     // ds_load_b64, conflict-free
        v2f_t b0 = *(const v2f_t*)&Wr0[k0];
        v2f_t b1 = *(const v2f_t*)&Wr1[k0];
        v2f_t b2 = *(const v2f_t*)&Wr2[k0];
        v2f_t b3 = *(const v2f_t*)&Wr3[k0];
        acc0 = __builtin_amdgcn_wmma_f32_16x16x4_f32(false, a, false, b0, (short)0, acc0, false, false);
        acc1 = __builtin_amdgcn_wmma_f32_16x16x4_f32(false, a, false, b1, (short)0, acc1, false, false);
        acc2 = __builtin_amdgcn_wmma_f32_16x16x4_f32(false, a, false, b2, (short)0, acc2, false, false);
        acc3 = __builtin_amdgcn_wmma_f32_16x16x4_f32(false, a, false, b3, (short)0, acc3, false, false);
    }

    if (!valid) return;
    // C/D layout: VGPR r -> row = r + (lane>=16 ? 8 : 0), col = t*16 + (lane&15)
    const int roff = (lane >> 4) * 8;
    const int c = lane & 15;
    v8f_t accs[4] = {acc0, acc1, acc2, acc3};
#pragma unroll
    for (int t = 0; t < 4; ++t) {
#pragma unroll
        for (int r = 0; r < 8; ++r) {
            int row = row_base + roff + r;
            if (row >= n_rows) continue;
            int col = t * 16 + c;
            float cv = accs[t][r];
            float o;
            if (mode == 0) {
                float hv = Atw[(roff + r) * APAD + col];   // h from LDS tile
                o = (1.0f - beta) * hv + beta * cv;
                o = o > 0.0f ? o : 0.0f;
            } else {
                o = cv + bias[col];
            }
            out[(size_t)row * HN + col] = o;
        }
    }
}

// ---------------------------------------------------------------- pool
__global__ __launch_bounds__(256) void poolinit_kernel(unsigned int* __restrict__ p, int n)
{
    int i = blockIdx.x * blockDim.x + threadIdx.x;
    if (i < n) p[i] = 0xFF800000u;  // -inf
}

__device__ inline void atomicMaxF(float* addr, float v)
{
    if (!(v == v)) return;
    if (v >= 0.0f) atomicMax((int*)addr, __float_as_int(v));
    else           atomicMin((unsigned int*)addr, __float_as_uint(v));
}

__global__ __launch_bounds__(256) void pool_kernel(
    const float* __restrict__ y, const int* __restrict__ batch,
    float* __restrict__ pooled, int n)
{
    int gid = blockIdx.x * blockDim.x + threadIdx.x;
    if (gid >= n * HN) return;
    int node = gid >> 6, c = gid & 63;
    int g = batch[node];
    atomicMaxF(&pooled[(size_t)g * HN + c], y[gid]);
}

// ---------------------------------------------------------------- head MLP (single block)
__global__ __launch_bounds__(256) void head_kernel(
    const float* __restrict__ pooled,
    const float* __restrict__ m1w, const float* __restrict__ m1b,
    const float* __restrict__ g1, const float* __restrict__ b1,
    const float* __restrict__ m2w, const float* __restrict__ m2b,
    const float* __restrict__ g2, const float* __restrict__ b2,
    const float* __restrict__ ow, const float* __restrict__ ob,
    float* __restrict__ out)
{
    __shared__ float P[GC * HN];
    __shared__ float Z[GC * HN];
    __shared__ float mu[HN], rs[HN];
    const int tid = threadIdx.x;

    for (int i = tid; i < GC * HN; i += 256) {
        float v = pooled[i];
        P[i] = (v == v && v >= -3.0e38f && v <= 3.0e38f) ? v : 0.0f;
    }
    __syncthreads();

    const float* mw[2] = {m1w, m2w};
    const float* mb[2] = {m1b, m2b};
    const float* bg[2] = {g1, g2};
    const float* bb[2] = {b1, b2};

    for (int layer = 0; layer < 2; ++layer) {
        const float* wL = mw[layer];
        const float* bL = mb[layer];
        for (int i = tid; i < GC * HN; i += 256) {
            int g = i >> 6, j = i & 63;
            float acc = bL[j];
            const float* prow = &P[g * HN];
#pragma unroll 8
            for (int k = 0; k < HN; ++k) acc = fmaf(prow[k], wL[k * HN + j], acc);
            Z[i] = acc;
        }
        __syncthreads();
        if (tid < HN) {
            float s = 0.0f, s2 = 0.0f;
            for (int g = 0; g < GC; ++g) { float v = Z[g * HN + tid]; s += v; s2 = fmaf(v, v, s2); }
            float mean = s * (1.0f / GC);
            float var = s2 * (1.0f / GC) - mean * mean;
            mu[tid] = mean;
            rs[tid] = rsqrtf(var + 1e-5f);
        }
        __syncthreads();
        for (int i = tid; i < GC * HN; i += 256) {
            int j = i & 63;
            float v = bg[layer][j] * (Z[i] - mu[j]) * rs[j] + bb[layer][j];
            P[i] = v > 0.0f ? v : 0.0f;
        }
        __syncthreads();
    }

    if (tid < GC) {
        float logits[10];
        float mx = -1e30f;
#pragma unroll
        for (int t = 0; t < 10; ++t) {
            float acc = ob[t];
            const float* prow = &P[tid * HN];
#pragma unroll 8
            for (int k = 0; k < HN; ++k) acc = fmaf(prow[k], ow[k * 10 + t], acc);
            logits[t] = acc;
            mx = fmaxf(mx, acc);
        }
        float se = 0.0f;
#pragma unroll
        for (int t = 0; t < 10; ++t) se += __expf(logits[t] - mx);
        float lse = mx + __logf(se);
#pragma unroll
        for (int t = 0; t < 10; ++t) out[tid * 10 + t] = logits[t] - lse;
    }
}

// ---------------------------------------------------------------- launch
extern "C" void kernel_launch(void* const* d_in, const int* in_sizes, int n_in,
                              void* d_out, int out_size, void* d_ws, size_t ws_size,
                              hipStream_t stream)
{
    const float* pos    = (const float*)d_in[0];
    const int*   ei     = (const int*)d_in[1];
    const int*   batch  = (const int*)d_in[2];
    const float* lin0_w = (const float*)d_in[3];
    const float* lin0_b = (const float*)d_in[4];
    const float* conv_w = (const float*)d_in[5];
    const float* lin1_w = (const float*)d_in[6];
    const float* lin1_b = (const float*)d_in[7];
    const float* mlp1_w = (const float*)d_in[8];
    const float* mlp1_b = (const float*)d_in[9];
    const float* bn1_g  = (const float*)d_in[10];
    const float* bn1_b  = (const float*)d_in[11];
    const float* mlp2_w = (const float*)d_in[12];
    const float* mlp2_b = (const float*)d_in[13];
    const float* bn2_g  = (const float*)d_in[14];
    const float* bn2_b  = (const float*)d_in[15];
    const float* out_w  = (const float*)d_in[16];
    const float* out_b  = (const float*)d_in[17];

    const int nN = in_sizes[2];          // 100000
    const int nE = in_sizes[1] / 2;      // 1200000

    float* x      = (float*)d_ws;
    float* x0     = x  + (size_t)nN * HN;
    float* h      = x0 + (size_t)nN * HN;
    float* pooled = h  + (size_t)nN * HN;

    const int elem_blocks = (nN * HN + 255) / 256;
    const int tiles = (nN + 15) / 16;
    const int gemm_blocks = (tiles + 7) / 8;

    lin0_kernel<<<elem_blocks, 256, 0, stream>>>(pos, lin0_w, lin0_b, x, x0, nN);

    for (int l = 0; l < LAYERS; ++l) {
        float beta = logf(0.5f / (float)(l + 1) + 1.0f);
        inith_kernel<<<(nN * 16 + 255) / 256, 256, 0, stream>>>(
            (const float4*)x0, (float4*)h, nN * 16);
        scatter_kernel<<<(nE * 16 + 255) / 256, 256, 0, stream>>>(
            x, ei, ei + nE, h, nE);
        gemm64_kernel<<<gemm_blocks, 256, 0, stream>>>(
            h, conv_w + (size_t)l * HN * HN, nullptr, x, nN, beta, 0);
    }

    // lin1: h = x @ lin1_w + lin1_b
    gemm64_kernel<<<gemm_blocks, 256, 0, stream>>>(x, lin1_w, lin1_b, h, nN, 0.0f, 1);

    poolinit_kernel<<<(GC * HN + 255) / 256, 256, 0, stream>>>((unsigned int*)pooled, GC * HN);
    pool_kernel<<<elem_blocks, 256, 0, stream>>>(h, batch, pooled, nN);

    head_kernel<<<1, 256, 0, stream>>>(pooled, mlp1_w, mlp1_b, bn1_g, bn1_b,
                                       mlp2_w, mlp2_b, bn2_g, bn2_b, out_w, out_b,
                                       (float*)d_out);
}